// Net_5523327943005
// MI455X (gfx1250) — compile-verified
//
#include <hip/hip_runtime.h>

typedef __attribute__((ext_vector_type(2))) float v2f;
typedef __attribute__((ext_vector_type(8))) float v8f;

#define S_EDGE 8   // edge feature dim
#define NT 5       // N-tiles (16 cols each) per wave in the WMMA GEMM

// ---------------------------------------------------------------------------
// Pack B (fi x 10*fo), columns laid out for the consumers:
//   c in [0, 8*fo):     o-major V:  o = c/8, s = c%8 -> We[(s*fi+f)*fo + o]
//     (so the edge kernel reads V[n, o*8 .. o*8+7] as two float4)
//   c in [8*fo, 9*fo):  Vb: be[f*fo + (c-8*fo)]
//   c in [9*fo,10*fo):  R:  root[f*fo + (c-9*fo)]
// ---------------------------------------------------------------------------
__global__ void pack_B(const float* __restrict__ We, const float* __restrict__ be,
                       const float* __restrict__ root, float* __restrict__ Bp,
                       int fi, int fo) {
    int idx = blockIdx.x * blockDim.x + threadIdx.x;
    int ncols = 10 * fo;
    int total = fi * ncols;
    if (idx >= total) return;
    int f = idx / ncols, c = idx % ncols;
    float v;
    if (c < 8 * fo) {
        int o = c >> 3, s = c & 7;                 // o-major
        v = We[((long)s * fi + f) * fo + o];
    } else if (c < 9 * fo) {
        v = be[(long)f * fo + (c - 8 * fo)];
    } else {
        v = root[(long)f * fo + (c - 9 * fo)];
    }
    Bp[(long)f * ncols + c] = v;
}

// ---------------------------------------------------------------------------
// C(M x ncols) = [relu](A)(M x K) * B(K x ncols), fp32 WMMA 16x16x4.
// One wave computes a 16 x (NT*16) strip: A panel held in registers across
// all NT tiles (NK = K/4 fragments, fully unrolled -> static VGPR indexing).
// Fragment layouts per ISA 7.12.2:
//   A (16x4): lanes 0-15 K=0,1 ; lanes 16-31 K=2,3 (M = lane&15)
//   B (4x16): lanes 0-15 K=0,1 ; lanes 16-31 K=2,3 (N = lane&15)
//   C/D     : VGPR j -> M=j (lanes 0-15), M=8+j (lanes 16-31)
// ---------------------------------------------------------------------------
template <int NK, bool RELU_IN>
__global__ __launch_bounds__(32)
void gemm_wmma_f32(const float* __restrict__ A, const float* __restrict__ B,
                   float* __restrict__ C, int Mrows, int ncols) {
    const int K    = NK * 4;
    const int lane = threadIdx.x & 31;
    const int half = lane >> 4;     // 0 or 1
    const int r16  = lane & 15;
    const int m0 = blockIdx.y << 4;
    const int n0 = (int)blockIdx.x * (NT * 16);

    int arow = m0 + r16;
    if (arow >= Mrows) arow = Mrows - 1;          // clamp (M % 16 == 0 here anyway)
    const float* Arow = A + (long)arow * K + (half << 1);

    // ---- A panel in registers (NK fragments) ----
    v2f afrag[NK];
#pragma unroll
    for (int kk = 0; kk < NK; ++kk) {
        float2 t = *(const float2*)(Arow + kk * 4);
        v2f a;
        a.x = RELU_IN ? fmaxf(t.x, 0.0f) : t.x;
        a.y = RELU_IN ? fmaxf(t.y, 0.0f) : t.y;
        afrag[kk] = a;
    }

    // ---- accumulate NT tiles ----
    v8f acc[NT];
#pragma unroll
    for (int t = 0; t < NT; ++t) acc[t] = (v8f){};

#pragma unroll
    for (int kk = 0; kk < NK; ++kk) {
        const int ka = (kk << 2) + (half << 1);
        const float* Bp0 = B + (long)ka * ncols + n0 + r16;
        const float* Bp1 = Bp0 + ncols;
#pragma unroll
        for (int t = 0; t < NT; ++t) {
            v2f b;
            b.x = Bp0[t * 16];
            b.y = Bp1[t * 16];
            acc[t] = __builtin_amdgcn_wmma_f32_16x16x4_f32(
                false, afrag[kk], false, b, (short)0, acc[t], false, false);
        }
    }

    // ---- store ----
    const int mbase = m0 + (half << 3);
#pragma unroll
    for (int t = 0; t < NT; ++t) {
        float* Crow = C + (long)mbase * ncols + n0 + t * 16 + r16;
#pragma unroll
        for (int j = 0; j < 8; ++j)
            if (mbase + j < Mrows) Crow[(long)j * ncols] = acc[t][j];
    }
}

// Scalar fallback for unexpected shapes (keeps the driver total-coverage safe).
__global__ void gemm_naive(const float* __restrict__ A, const float* __restrict__ B,
                           float* __restrict__ C, int Mrows, int K, int ncols, int relu_in) {
    long idx = (long)blockIdx.x * blockDim.x + threadIdx.x;
    if (idx >= (long)Mrows * ncols) return;
    long m = idx / ncols;
    int  c = (int)(idx - m * ncols);
    float acc = 0.0f;
    for (int k = 0; k < K; ++k) {
        float a = A[m * K + k];
        if (relu_in) a = fmaxf(a, 0.0f);
        acc += a * B[(long)k * ncols + c];
    }
    C[idx] = acc;
}

// ---------------------------------------------------------------------------
// agg[n,o] = U[n, 9*fo + o] (root contribution) + b[o]
// ---------------------------------------------------------------------------
__global__ void node_init(const float* __restrict__ U, const float* __restrict__ bias,
                          float* __restrict__ agg, int fo, long total) {
    long idx = (long)blockIdx.x * blockDim.x + threadIdx.x;
    if (idx >= total) return;
    long n = idx / fo;
    int  o = (int)(idx - n * fo);
    agg[idx] = U[n * (long)(10 * fo) + 9 * fo + o] + bias[o];
}

// ---------------------------------------------------------------------------
// Per-(edge, o):  msg = Vb[src,o] + sum_s e[e,s] * V[src, o*8+s]
// V is o-major -> the 8 weights are two aligned float4 (global_load_b128).
// U row stride 10*fo floats (1600/960 B, 16B-aligned). V <= 32 MB -> L2 resident.
// ---------------------------------------------------------------------------
__global__ void ecc_edge(const int* __restrict__ edge_index, const float* __restrict__ e,
                         const float* __restrict__ U, int fo,
                         float* __restrict__ agg, long total) {
    long idx = (long)blockIdx.x * blockDim.x + threadIdx.x;
    if (idx >= total) return;
    long eidx = idx / fo;
    int  o    = (int)(idx - eidx * fo);

    int2 st = *(const int2*)(edge_index + eidx * 2);   // {src, tgt}
    const float4* ev = (const float4*)(e + eidx * S_EDGE);
    float4 e0 = ev[0], e1 = ev[1];

    const float* Vn = U + (long)st.x * (10 * fo);
    float4 v0 = *(const float4*)(Vn + o * 8);
    float4 v1 = *(const float4*)(Vn + o * 8 + 4);

    float m = Vn[8 * fo + o];                          // Vb
    m += e0.x * v0.x + e0.y * v0.y + e0.z * v0.z + e0.w * v0.w;
    m += e1.x * v1.x + e1.y * v1.y + e1.z * v1.z + e1.w * v1.w;

    unsafeAtomicAdd(&agg[(long)st.y * fo + o], m);
}

// ---------------------------------------------------------------------------
// Column sums with ReLU fold: g[o] = sum_n max(h[n,o], 0)
// One block per column -> deterministic.
// ---------------------------------------------------------------------------
__global__ void col_sum_relu(const float* __restrict__ h, int n_rows, int fo,
                             float* __restrict__ g) {
    __shared__ float red[256];
    int o = blockIdx.x;
    float acc = 0.0f;
    for (int n = threadIdx.x; n < n_rows; n += blockDim.x)
        acc += fmaxf(h[(long)n * fo + o], 0.0f);
    red[threadIdx.x] = acc;
    __syncthreads();
    for (int s = 128; s > 0; s >>= 1) {
        if ((int)threadIdx.x < s) red[threadIdx.x] += red[threadIdx.x + s];
        __syncthreads();
    }
    if (threadIdx.x == 0) g[o] = red[0];
}

// ---------------------------------------------------------------------------
// Tiny MLP head, single block. dims: 24->96->256->768->512->64->1
// ---------------------------------------------------------------------------
struct MlpParams {
    const float* W[6];
    const float* b[6];
};

__global__ void mlp_head(const float* __restrict__ g, MlpParams p, float* __restrict__ out) {
    __shared__ float bufA[768];
    __shared__ float bufB[768];
    const int dims[7] = {24, 96, 256, 768, 512, 64, 1};
    float* cur = bufA;
    float* nxt = bufB;
    for (int i = threadIdx.x; i < 24; i += blockDim.x) cur[i] = g[i];
    for (int l = 0; l < 6; ++l) {
        __syncthreads();
        const int fin = dims[l], fout = dims[l + 1];
        const float* W  = p.W[l];
        const float* bb = p.b[l];
        for (int o = threadIdx.x; o < fout; o += blockDim.x) {
            float acc = bb[o];
            for (int i = 0; i < fin; ++i) acc += cur[i] * W[(long)i * fout + o];
            nxt[o] = (l < 5) ? fmaxf(acc, 0.0f) : acc;
        }
        float* t = cur; cur = nxt; nxt = t;
    }
    __syncthreads();
    if (threadIdx.x == 0) out[0] = cur[0];
}

// ---------------------------------------------------------------------------
// Host-side orchestration
// ---------------------------------------------------------------------------
static inline long cdiv(long a, long b) { return (a + b - 1) / b; }

static void launch_gemm(hipStream_t stream, const float* A, const float* B, float* C,
                        int N, int K, int ncols, bool relu_in) {
    const bool wmma_ok = (ncols % (NT * 16) == 0) && (K == 16 || K == 40 || K == 24);
    if (wmma_ok) {
        dim3 grid((unsigned)(ncols / (NT * 16)), (unsigned)cdiv(N, 16));
        if (K == 16) {
            if (relu_in) gemm_wmma_f32<4,  true ><<<grid, 32, 0, stream>>>(A, B, C, N, ncols);
            else         gemm_wmma_f32<4,  false><<<grid, 32, 0, stream>>>(A, B, C, N, ncols);
        } else if (K == 40) {
            if (relu_in) gemm_wmma_f32<10, true ><<<grid, 32, 0, stream>>>(A, B, C, N, ncols);
            else         gemm_wmma_f32<10, false><<<grid, 32, 0, stream>>>(A, B, C, N, ncols);
        } else {
            if (relu_in) gemm_wmma_f32<6,  true ><<<grid, 32, 0, stream>>>(A, B, C, N, ncols);
            else         gemm_wmma_f32<6,  false><<<grid, 32, 0, stream>>>(A, B, C, N, ncols);
        }
    } else {
        long total = (long)N * ncols;
        gemm_naive<<<(unsigned)cdiv(total, 256), 256, 0, stream>>>(A, B, C, N, K, ncols,
                                                                   relu_in ? 1 : 0);
    }
}

static void run_ecc_layer(hipStream_t stream,
                          const float* h_in, int fi, int fo, int N, long E, bool relu_in,
                          const float* We, const float* be, const float* root, const float* b,
                          const int* edge_index, const float* e,
                          float* Bpack, float* U, float* h_out) {
    const int ncols = 10 * fo;
    // 1) pack B = [We' (o-major) | be_mat | root]
    {
        long total = (long)fi * ncols;
        pack_B<<<(unsigned)cdiv(total, 256), 256, 0, stream>>>(We, be, root, Bpack, fi, fo);
    }
    // 2) U = relu?(h_in) @ Bpack   (fp32 WMMA)
    launch_gemm(stream, h_in, Bpack, U, N, fi, ncols, relu_in);
    // 3) agg init = root part + bias
    {
        long total = (long)N * fo;
        node_init<<<(unsigned)cdiv(total, 256), 256, 0, stream>>>(U, b, h_out, fo, total);
    }
    // 4) edge scatter (hardware f32 atomics)
    {
        long total = E * fo;
        ecc_edge<<<(unsigned)cdiv(total, 256), 256, 0, stream>>>(edge_index, e, U, fo, h_out, total);
    }
    // (ReLU folded into the next consumer: GEMM A-load / pooling)
}

extern "C" void kernel_launch(void* const* d_in, const int* in_sizes, int n_in,
                              void* d_out, int out_size, void* d_ws, size_t ws_size,
                              hipStream_t stream) {
    // ---- inputs (setup_inputs dict order) ----
    const float* x          = (const float*)d_in[0];
    const int*   edge_index = (const int*)  d_in[1];
    const float* e          = (const float*)d_in[2];
    const float* We1  = (const float*)d_in[3];
    const float* be1  = (const float*)d_in[4];
    const float* root1= (const float*)d_in[5];
    const float* b1   = (const float*)d_in[6];
    const float* We2  = (const float*)d_in[7];
    const float* be2  = (const float*)d_in[8];
    const float* root2= (const float*)d_in[9];
    const float* b2   = (const float*)d_in[10];
    const float* We3  = (const float*)d_in[11];
    const float* be3  = (const float*)d_in[12];
    const float* root3= (const float*)d_in[13];
    const float* b3   = (const float*)d_in[14];

    MlpParams mp;
    for (int l = 0; l < 6; ++l) {
        mp.W[l] = (const float*)d_in[15 + 2 * l];
        mp.b[l] = (const float*)d_in[16 + 2 * l];
    }

    // ---- shapes (derived generically) ----
    const int F  = 16;
    const int N  = in_sizes[0] / F;
    const long E = in_sizes[1] / 2;
    const int fo1 = in_sizes[6];                 // 40
    const int fo2 = in_sizes[10];                // 24
    const int fo3 = in_sizes[14];                // 24
    const int fi1 = in_sizes[5] / fo1;           // 16
    const int fi2 = in_sizes[9] / fo2;           // 40
    const int fi3 = in_sizes[13] / fo3;          // 24

    // ---- workspace layout ----
    char* ws = (char*)d_ws;
    size_t off = 0;
    auto alloc = [&](size_t bytes) { char* p = ws + off; off = (off + bytes + 255) & ~(size_t)255; return p; };
    float* Bpack = (float*)alloc(sizeof(float) * 16384);                 // >= fi*10*fo max (9600)
    float* U     = (float*)alloc(sizeof(float) * (size_t)N * 10 * fo1);  // 32 MB (largest layer)
    float* h1    = (float*)alloc(sizeof(float) * (size_t)N * fo1);
    float* h2    = (float*)alloc(sizeof(float) * (size_t)N * fo2);
    float* h3    = (float*)alloc(sizeof(float) * (size_t)N * fo3);
    float* g     = (float*)alloc(sizeof(float) * 32);
    (void)ws_size;

    // ---- 3 ECC layers (ReLU of layer i folded into consumer of layer i) ----
    run_ecc_layer(stream, x,  fi1, fo1, N, E, /*relu_in=*/false, We1, be1, root1, b1,
                  edge_index, e, Bpack, U, h1);
    run_ecc_layer(stream, h1, fi2, fo2, N, E, /*relu_in=*/true,  We2, be2, root2, b2,
                  edge_index, e, Bpack, U, h2);
    run_ecc_layer(stream, h2, fi3, fo3, N, E, /*relu_in=*/true,  We3, be3, root3, b3,
                  edge_index, e, Bpack, U, h3);

    // ---- pooling with ReLU fold: g[o] = sum_n relu(h3[n,o]) ----
    col_sum_relu<<<(unsigned)fo3, 256, 0, stream>>>(h3, N, fo3, g);

    // ---- MLP head -> single scalar ----
    mlp_head<<<1, 256, 0, stream>>>(g, mp, (float*)d_out);
}